// MetaLEAPPredictor_74208444940496
// MI455X (gfx1250) — compile-verified
//
#include <hip/hip_runtime.h>
#include <hip/hip_bf16.h>

// ---------------------------------------------------------------------------
// MetaLEAP edge predictor, factorized:
//   s[e] = sf[e] . (Wa^T x[row]) + sf[e] . (Wb^T x[col]) + x[row].c_a + x[col].c_b
// where Wa = psi_w[:64,:], Wb = psi_w[64:,:], c = psi_b + delta_w[li] + u[li].
//
// Phase 1: build Bmat [64 x 16]: cols 0-3 = Wa, 4-7 = Wb, 8 = c_a, 9 = c_b,
//          10-15 = 0.
// Phase 2 (WMMA fp32): proj[N x 16] = x[N x 64] @ Bmat, via
//          v_wmma_f32_16x16x4_f32 (exact fp32 math, 16 K-steps of 4).
// Phase 3: per-edge: gather 2 x (float4 + float) from proj, dot with sf,
//          leaky_relu, scale by gamma_h[li], write 8 contiguous outputs.
// ---------------------------------------------------------------------------

typedef __attribute__((ext_vector_type(2))) float v2f;
typedef __attribute__((ext_vector_type(8))) float v8f;

#define NEG_SLOPE 0.01f

// ---------------- Phase 1: tiny B-matrix prep (64 threads) -----------------
__global__ void prep_bmat_kernel(const float* __restrict__ psi_w,   // [128,4]
                                 const float* __restrict__ psi_b,   // [128]
                                 const float* __restrict__ delta_w, // [L-1,128]
                                 const float* __restrict__ u,       // [L-1,128]
                                 const int*   __restrict__ layer_idx,
                                 float* __restrict__ Bmat)          // [64,16]
{
    const int k = threadIdx.x;            // channel 0..63
    if (k >= 64) return;
    const int li = layer_idx[0];          // small value; int32 view of int64 is fine (LE)
    const float* dw = delta_w + (long)li * 128;
    const float* uu = u       + (long)li * 128;
    float* row = Bmat + k * 16;
#pragma unroll
    for (int j = 0; j < 4; ++j) {
        row[j]     = psi_w[k * 4 + j];          // Wa[k][j]
        row[4 + j] = psi_w[(64 + k) * 4 + j];   // Wb[k][j]
    }
    row[8] = psi_b[k]      + dw[k]      + uu[k];        // c_a[k]
    row[9] = psi_b[64 + k] + dw[64 + k] + uu[64 + k];   // c_b[k]
#pragma unroll
    for (int j = 10; j < 16; ++j) row[j] = 0.0f;
}

// ---------------- Phase 2: node projection via fp32 WMMA -------------------
// One wave32 per 16 nodes. proj[n, 0:16] = x[n, 0:64] @ Bmat[64, 16].
// A layout (ISA 16x4 f32): lane L, vgpr v -> A[M = L%16][K = (L/16)*2 + v]
// B layout (assumed symmetric): lane L, vgpr v -> B[K = (L/16)*2 + v][N = L%16]
// D layout (ISA 16x16 f32): lane L, vgpr r -> D[M = r + 8*(L/16)][N = L%16]
__global__ void node_proj_wmma_kernel(const float* __restrict__ x,    // [N,64]
                                      const float* __restrict__ Bmat, // [64,16]
                                      float* __restrict__ proj,       // [N,16]
                                      int nNodes)
{
    const int lane  = threadIdx.x & 31;
    const int wib   = threadIdx.x >> 5;
    const long wave = (long)blockIdx.x * (blockDim.x >> 5) + wib;
    const int base  = (int)(wave * 16);
    if (base >= nNodes) return;           // wave-uniform: EXEC stays all-ones

    const int n   = lane & 15;            // N-col (and A-row) index
    const int hi  = lane >> 4;            // lane-half
    const int klo = hi * 2;               // this lane's K sub-offset

    // Preload B fragments for all 16 K-steps (Bmat is 4KB, cache-resident).
    v2f bfrag[16];
#pragma unroll
    for (int t = 0; t < 16; ++t) {
        const int k0 = 4 * t + klo;
        v2f b; b.x = Bmat[k0 * 16 + n]; b.y = Bmat[(k0 + 1) * 16 + n];
        bfrag[t] = b;
    }

    // Clamp row so tail waves (generic N) never read OOB; EXEC stays full.
    const int mrow = min(base + n, nNodes - 1);
    const float* xrow = x + (long)mrow * 64;

    v8f acc = {};
#pragma unroll
    for (int t = 0; t < 16; ++t) {
        const int k0 = 4 * t + klo;               // even -> 8B-aligned b64 load
        const v2f a = *(const v2f*)(xrow + k0);
        acc = __builtin_amdgcn_wmma_f32_16x16x4_f32(
            /*neg_a=*/false, a, /*neg_b=*/false, bfrag[t],
            /*c_mod=*/(short)0, acc, /*reuse_a=*/false, /*reuse_b=*/false);
    }

    // Scatter D: lane L vgpr r holds proj[base + r + 8*hi][n].
#pragma unroll
    for (int r = 0; r < 8; ++r) {
        const int m = base + r + 8 * hi;
        if (m < nNodes) proj[(long)m * 16 + n] = acc[r];
    }
}

// ---------------- Phase 3: per-edge assembly -------------------------------
__global__ void edge_kernel(const long long* __restrict__ edge_index, // [2,E] i64
                            const float* __restrict__ sf,             // [E,4]
                            const float* __restrict__ proj,           // [N,16]
                            const float* __restrict__ gamma_h,        // [L-1,8]
                            const int*   __restrict__ layer_idx,
                            float* __restrict__ out,                  // [E,8]
                            int nEdges)
{
    const long e = (long)blockIdx.x * blockDim.x + threadIdx.x;
    if (e >= nEdges) return;
    const int li = layer_idx[0];

    const long row = edge_index[e];
    const long col = edge_index[(long)nEdges + e];

    const float4 sfe = *(const float4*)(sf + e * 4);
    const float* pr = proj + row * 16;
    const float* pc = proj + col * 16;
    const float4 ar = *(const float4*)(pr);      // Wa^T x[row]
    const float4 bc = *(const float4*)(pc + 4);  // Wb^T x[col]
    const float  ca = pr[8];                     // x[row].c_a
    const float  cb = pc[9];                     // x[col].c_b

    float s = sfe.x * (ar.x + bc.x) + sfe.y * (ar.y + bc.y)
            + sfe.z * (ar.z + bc.z) + sfe.w * (ar.w + bc.w)
            + ca + cb;
    const float base = (s >= 0.0f) ? s : NEG_SLOPE * s;

    const float* g = gamma_h + (long)li * 8;
    float4 o0, o1;
    o0.x = g[0] * base; o0.y = g[1] * base; o0.z = g[2] * base; o0.w = g[3] * base;
    o1.x = g[4] * base; o1.y = g[5] * base; o1.z = g[6] * base; o1.w = g[7] * base;
    *(float4*)(out + e * 8)     = o0;            // 32B-aligned contiguous row
    *(float4*)(out + e * 8 + 4) = o1;
}

// ---------------------------------------------------------------------------
extern "C" void kernel_launch(void* const* d_in, const int* in_sizes, int n_in,
                              void* d_out, int out_size, void* d_ws, size_t ws_size,
                              hipStream_t stream)
{
    const float*     x       = (const float*)d_in[0];      // [N,64] f32
    const long long* ei      = (const long long*)d_in[1];  // [2,E] i64
    const float*     sf      = (const float*)d_in[2];      // [E,4] f32
    const int*       li      = (const int*)d_in[3];        // scalar
    const float*     psi_w   = (const float*)d_in[4];      // [128,4]
    const float*     psi_b   = (const float*)d_in[5];      // [128]
    const float*     delta_w = (const float*)d_in[6];      // [2,128]
    const float*     u       = (const float*)d_in[7];      // [2,128]
    const float*     gamma_h = (const float*)d_in[8];      // [2,8]
    float*           out     = (float*)d_out;              // [E,8]

    const int nNodes = in_sizes[0] / 64;
    const int nEdges = in_sizes[2] / 4;

    // ws layout: [0,4096) Bmat (64x16 f32) ; [4096, ...) proj (N x 16 f32)
    float* Bmat = (float*)d_ws;
    float* proj = (float*)((char*)d_ws + 4096);

    prep_bmat_kernel<<<1, 64, 0, stream>>>(psi_w, psi_b, delta_w, u, li, Bmat);

    const int waves   = (nNodes + 15) / 16;
    const int wpb     = 8;                       // 8 wave32 = 256 threads/block
    const int nblocks = (waves + wpb - 1) / wpb;
    node_proj_wmma_kernel<<<nblocks, wpb * 32, 0, stream>>>(x, Bmat, proj, nNodes);

    const int eblocks = (nEdges + 255) / 256;
    edge_kernel<<<eblocks, 256, 0, stream>>>(ei, sf, proj, gamma_h, li, out, nEdges);
}